// GNNModel_59493886984415
// MI455X (gfx1250) — compile-verified
//
#include <hip/hip_runtime.h>
#include <cstdint>

// ---------------------------------------------------------------------------
// GraphSAGE (hetero, 2-layer) for MI455X / gfx1250.
//
// Roofline: ~20.6 GFLOP of fp32 GEMM vs ~6 GB of edge-gather/scatter traffic
// -> memory/atomic bound at 23.3 TB/s (~260us floor). Therefore we keep full
// fp32 precision and use V_WMMA_F32_16X16X4_F32 for all dense transforms.
// Scatter destinations (51MB + 26MB) are resident in the 192MB L2, so the
// ~512M global_atomic_add_f32 ops are serviced by L2 atomic units.
// Weight staging uses GLOBAL_LOAD_ASYNC_TO_LDS_B128 (ASYNCcnt) so the
// 128KB/block weight copy bypasses VGPRs and overlaps the wave prologue.
// ---------------------------------------------------------------------------

#define NU_N   100000
#define NG_N   50000
#define HD     128
#define NE_N   1000000
#define NEL_N  500000
#define NGEN_N 74

// Set to 0 to fall back to synchronous VGPR staging if the assembler rejects
// the async mnemonics.
#define USE_ASYNC_LDS 1

typedef __attribute__((ext_vector_type(2))) float v2f;
typedef __attribute__((ext_vector_type(8))) float v8f;

// D = A(16x4 fp32) * B(4x16 fp32) + C(16x16 fp32), full wave32 cooperative.
__device__ __forceinline__ v8f wmma_f32(v2f a, v2f b, v8f c) {
  return __builtin_amdgcn_wmma_f32_16x16x4_f32(
      /*neg_a=*/false, a, /*neg_b=*/false, b,
      /*c_mod=*/(short)0, c, /*reuse_a=*/false, /*reuse_b=*/false);
}

// ---------------------------------------------------------------------------
__global__ void zero_kernel(float* __restrict__ p, size_t n) {
  size_t i = (size_t)blockIdx.x * blockDim.x + threadIdx.x;
  size_t stride = (size_t)gridDim.x * blockDim.x;
  for (; i < n; i += stride) p[i] = 0.0f;
}

// In-degree counts for both node types (shared by both layers).
__global__ void degree_kernel(const int* __restrict__ esrc,
                              const int* __restrict__ edst,
                              float* __restrict__ cnt_u,
                              float* __restrict__ cnt_g) {
  int t = blockIdx.x * blockDim.x + threadIdx.x;
  if (t < NE_N) {
    atomicAdd(&cnt_u[esrc[t]], 1.0f);
    atomicAdd(&cnt_g[edst[t]], 1.0f);
  }
}

// One wave32 per edge; lane l carries cols [4l,4l+3] as a float4.
// msum[sidx[e]] += xsrc[gidx[e]]  (f32 atomics, L2-resident destination).
__global__ void scatter_sum_kernel(const float* __restrict__ xsrc,
                                   const int* __restrict__ gidx,
                                   const int* __restrict__ sidx,
                                   float* __restrict__ msum, int ne) {
  int lane = threadIdx.x & 31;
  int wid = (blockIdx.x * blockDim.x + threadIdx.x) >> 5;
  int nw = (gridDim.x * blockDim.x) >> 5;
  for (int e = wid; e < ne; e += nw) {
    __builtin_prefetch(&gidx[e + 4096], 0, 0);   // global_prefetch_b8
    __builtin_prefetch(&sidx[e + 4096], 0, 0);
    int s = gidx[e];
    int d = sidx[e];
    const float4 v = ((const float4*)(xsrc + (size_t)s * HD))[lane];
    float* drow = msum + (size_t)d * HD + lane * 4;
    atomicAdd(drow + 0, v.x);
    atomicAdd(drow + 1, v.y);
    atomicAdd(drow + 2, v.z);
    atomicAdd(drow + 3, v.w);
  }
}

// x_game = game_x[NG,74] @ W[74,128] + b + game_emb.  WMMA fp32, K padded.
// A-frag layout: lane half h holds K = 2h, 2h+1 for row (lane%16).
__global__ __launch_bounds__(256) void game_lin_kernel(
    const float* __restrict__ gx, const float* __restrict__ w,
    const float* __restrict__ bias, const float* __restrict__ emb,
    float* __restrict__ out) {
  int wave = threadIdx.x >> 5;
  int lane = threadIdx.x & 31;
  int tile = blockIdx.x * 8 + wave;           // (NG/16)*8 tiles total
  int mt = tile >> 3, nt = tile & 7;
  int row0 = mt * 16, col0 = nt * 16;
  int h = lane >> 4, lm = lane & 15;
  int col = col0 + lm;

  v8f acc = {};
  const float* arow = gx + (size_t)(row0 + lm) * NGEN_N;
  for (int k = 0; k < NGEN_N; k += 4) {
    int k0 = k + 2 * h;
    v2f a, bf;
    a.x  = (k0     < NGEN_N) ? arow[k0]             : 0.0f;
    a.y  = (k0 + 1 < NGEN_N) ? arow[k0 + 1]         : 0.0f;
    bf.x = (k0     < NGEN_N) ? w[(size_t)k0 * HD + col]       : 0.0f;
    bf.y = (k0 + 1 < NGEN_N) ? w[(size_t)(k0 + 1) * HD + col] : 0.0f;
    acc = wmma_f32(a, bf, acc);
  }
  float bv = bias[col];
#pragma unroll
  for (int v = 0; v < 8; ++v) {           // C layout: M = v + 8h, N = lane%16
    int row = row0 + v + 8 * h;
    out[(size_t)row * HD + col] = acc[v] + bv + emb[(size_t)row * HD + col];
  }
}

// out = relu?( (msum/cnt) @ Wl + bl + xdst @ Wr ).
// Both 128x128 weights staged row-major into LDS via async b128 copies; each
// wave owns a 16-row strip and all 8 column tiles (64 acc VGPRs), issuing
// 16 v_wmma_f32_16x16x4_f32 per K-step.
__global__ __launch_bounds__(256) void sage_combine_kernel(
    const float* __restrict__ msum, const float* __restrict__ cnt,
    const float* __restrict__ xdst,
    const float* __restrict__ wl, const float* __restrict__ bl,
    const float* __restrict__ wr, float* __restrict__ out,
    int ntiles_m, int relu_flag) {
  extern __shared__ float lds[];
  float* Wl = lds;              // row-major copy: Wl[k*HD + c] = wl[k*HD + c]
  float* Wr = lds + HD * HD;

#if USE_ASYNC_LDS
  // 16 async b128 copies per thread per matrix; LDS dest addr == byte offset
  // into the dynamic LDS segment (no static LDS in this kernel -> base 0).
  for (int i = threadIdx.x; i < (HD * HD) / 4; i += blockDim.x) {
    unsigned la = (unsigned)(i * 16);
    unsigned lb = la + (unsigned)(HD * HD * 4);
    uint64_t ga = (uint64_t)(uintptr_t)(wl + (size_t)i * 4);
    uint64_t gb = (uint64_t)(uintptr_t)(wr + (size_t)i * 4);
    asm volatile("global_load_async_to_lds_b128 %0, %1, off"
                 :: "v"(la), "v"(ga) : "memory");
    asm volatile("global_load_async_to_lds_b128 %0, %1, off"
                 :: "v"(lb), "v"(gb) : "memory");
  }
  asm volatile("s_wait_asynccnt 0" ::: "memory");
#else
  for (int i = threadIdx.x; i < (HD * HD) / 4; i += blockDim.x) {
    ((float4*)Wl)[i] = ((const float4*)wl)[i];
    ((float4*)Wr)[i] = ((const float4*)wr)[i];
  }
#endif
  __syncthreads();

  int wave = threadIdx.x >> 5;
  int lane = threadIdx.x & 31;
  int mt = blockIdx.x * 8 + wave;
  if (mt >= ntiles_m) return;               // wave-uniform: EXEC stays all-1s
  int h = lane >> 4, lm = lane & 15;
  int row0 = mt * 16;
  int arow = row0 + lm;                     // only A-row this lane touches
  float rcp = 1.0f / fmaxf(cnt[arow], 1.0f);

  v8f acc[8] = {};
  const float* msrow = msum + (size_t)arow * HD;
  const float* xrow  = xdst + (size_t)arow * HD;

  for (int k = 0; k < HD; k += 4) {
    int k0 = k + 2 * h;
    v2f am, ax;
    am.x = msrow[k0] * rcp;  am.y = msrow[k0 + 1] * rcp;
    ax.x = xrow[k0];         ax.y = xrow[k0 + 1];
    const float* wl0 = Wl + (size_t)k0 * HD;        // B row k0
    const float* wl1 = Wl + (size_t)(k0 + 1) * HD;  // B row k0+1
    const float* wr0 = Wr + (size_t)k0 * HD;
    const float* wr1 = Wr + (size_t)(k0 + 1) * HD;
#pragma unroll
    for (int n = 0; n < 8; ++n) {
      int col = n * 16 + lm;
      v2f blf, brf;
      blf.x = wl0[col];  blf.y = wl1[col];
      brf.x = wr0[col];  brf.y = wr1[col];
      acc[n] = wmma_f32(am, blf, acc[n]);   // mean @ Wl
      acc[n] = wmma_f32(ax, brf, acc[n]);   // x_dst @ Wr
    }
  }

#pragma unroll
  for (int n = 0; n < 8; ++n) {
    int col = n * 16 + lm;
    float bv = bl[col];
#pragma unroll
    for (int v = 0; v < 8; ++v) {
      int row = row0 + v + 8 * h;
      float val = acc[n][v] + bv;
      if (relu_flag) val = fmaxf(val, 0.0f);
      out[(size_t)row * HD + col] = val;
    }
  }
}

// out[e] = dot(u2[el_src[e]], g2[el_dst[e]]) : wave per edge, shfl reduction.
__global__ void link_score_kernel(const float* __restrict__ u2,
                                  const float* __restrict__ g2,
                                  const int* __restrict__ els,
                                  const int* __restrict__ eld,
                                  float* __restrict__ out, int ne) {
  int lane = threadIdx.x & 31;
  int wid = (blockIdx.x * blockDim.x + threadIdx.x) >> 5;
  int nw = (gridDim.x * blockDim.x) >> 5;
  for (int e = wid; e < ne; e += nw) {
    const float4 a = ((const float4*)(u2 + (size_t)els[e] * HD))[lane];
    const float4 b = ((const float4*)(g2 + (size_t)eld[e] * HD))[lane];
    float s = a.x * b.x + a.y * b.y + a.z * b.z + a.w * b.w;
#pragma unroll
    for (int off = 16; off > 0; off >>= 1) s += __shfl_xor(s, off, 32);
    if (lane == 0) out[e] = s;
  }
}

// ---------------------------------------------------------------------------
extern "C" void kernel_launch(void* const* d_in, const int* in_sizes, int n_in,
                              void* d_out, int out_size, void* d_ws, size_t ws_size,
                              hipStream_t stream) {
  (void)in_sizes; (void)n_in; (void)out_size; (void)ws_size;
  // d_in[0]=user_node_id, d_in[1]=game_node_id (both identity -> unused)
  const float* game_x     = (const float*)d_in[2];
  const int*   edge_src   = (const int*)d_in[3];
  const int*   edge_dst   = (const int*)d_in[4];
  const int*   el_src     = (const int*)d_in[5];
  const int*   el_dst     = (const int*)d_in[6];
  const float* user_emb   = (const float*)d_in[7];
  const float* game_emb   = (const float*)d_in[8];
  const float* game_lin_w = (const float*)d_in[9];
  const float* game_lin_b = (const float*)d_in[10];
  const float* w1l_o = (const float*)d_in[11];
  const float* b1l_o = (const float*)d_in[12];
  const float* w1r_o = (const float*)d_in[13];
  const float* w1l_r = (const float*)d_in[14];
  const float* b1l_r = (const float*)d_in[15];
  const float* w1r_r = (const float*)d_in[16];
  const float* w2l_o = (const float*)d_in[17];
  const float* b2l_o = (const float*)d_in[18];
  const float* w2r_o = (const float*)d_in[19];
  const float* w2l_r = (const float*)d_in[20];
  const float* b2l_r = (const float*)d_in[21];
  const float* w2r_r = (const float*)d_in[22];

  // Workspace layout (~257 MB). cnt_u/cnt_g first (contiguous, one zero pass);
  // all feature-row buffers start at multiples of 4 floats -> float4 aligned.
  float* ws = (float*)d_ws;
  size_t off = 0;
  float* cnt_u = ws + off; off += NU_N;
  float* cnt_g = ws + off; off += NG_N;
  float* xg    = ws + off; off += (size_t)NG_N * HD;   // x_game
  float* u1    = ws + off; off += (size_t)NU_N * HD;
  float* g1    = ws + off; off += (size_t)NG_N * HD;
  float* u2b   = ws + off; off += (size_t)NU_N * HD;
  float* g2b   = ws + off; off += (size_t)NG_N * HD;
  float* msu   = ws + off; off += (size_t)NU_N * HD;   // user-side segment sum
  float* msg   = ws + off; off += (size_t)NG_N * HD;   // game-side segment sum

  const size_t ldsBytes = (size_t)2 * HD * HD * sizeof(float);  // 128 KB

  // Degrees (computed once, reused by both layers).
  zero_kernel<<<1024, 256, 0, stream>>>(cnt_u, (size_t)(NU_N + NG_N));
  degree_kernel<<<(NE_N + 255) / 256, 256, 0, stream>>>(edge_src, edge_dst,
                                                        cnt_u, cnt_g);

  // Game feature projection (WMMA fp32): x_game = game_x@W + b + game_emb.
  game_lin_kernel<<<NG_N / 16, 256, 0, stream>>>(game_x, game_lin_w,
                                                 game_lin_b, game_emb, xg);

  // ---- Layer 1 ----
  zero_kernel<<<4096, 256, 0, stream>>>(msu, (size_t)NU_N * HD);
  scatter_sum_kernel<<<NE_N / 8, 256, 0, stream>>>(xg, edge_dst, edge_src,
                                                   msu, NE_N);
  sage_combine_kernel<<<(NU_N / 16 + 7) / 8, 256, ldsBytes, stream>>>(
      msu, cnt_u, user_emb, w1l_r, b1l_r, w1r_r, u1, NU_N / 16, 1);

  zero_kernel<<<4096, 256, 0, stream>>>(msg, (size_t)NG_N * HD);
  scatter_sum_kernel<<<NE_N / 8, 256, 0, stream>>>(user_emb, edge_src, edge_dst,
                                                   msg, NE_N);
  sage_combine_kernel<<<(NG_N / 16 + 7) / 8, 256, ldsBytes, stream>>>(
      msg, cnt_g, xg, w1l_o, b1l_o, w1r_o, g1, NG_N / 16, 1);

  // ---- Layer 2 (no activation) ----
  zero_kernel<<<4096, 256, 0, stream>>>(msu, (size_t)NU_N * HD);
  scatter_sum_kernel<<<NE_N / 8, 256, 0, stream>>>(g1, edge_dst, edge_src,
                                                   msu, NE_N);
  sage_combine_kernel<<<(NU_N / 16 + 7) / 8, 256, ldsBytes, stream>>>(
      msu, cnt_u, u1, w2l_r, b2l_r, w2r_r, u2b, NU_N / 16, 0);

  zero_kernel<<<4096, 256, 0, stream>>>(msg, (size_t)NG_N * HD);
  scatter_sum_kernel<<<NE_N / 8, 256, 0, stream>>>(u1, edge_src, edge_dst,
                                                   msg, NE_N);
  sage_combine_kernel<<<(NG_N / 16 + 7) / 8, 256, ldsBytes, stream>>>(
      msg, cnt_g, g1, w2l_o, b2l_o, w2r_o, g2b, NG_N / 16, 0);

  // Link classifier.
  link_score_kernel<<<NEL_N / 8, 256, 0, stream>>>(u2b, g2b, el_src, el_dst,
                                                   (float*)d_out, NEL_N);
}